// GATModel_88613765251622
// MI455X (gfx1250) — compile-verified
//
#include <hip/hip_runtime.h>
#include <math.h>

#define NN 100000
#define EE 1600000
#define ETOT (EE + NN)      // edges + self loops
#define NTILES (NN / 16)    // 6250 exact
#define NEG 0.2f

typedef __attribute__((ext_vector_type(2))) float v2f;
typedef __attribute__((ext_vector_type(8))) float v8f;

__device__ __forceinline__ float leaky(float v) { return v > 0.f ? v : NEG * v; }

// exact float atomic max via monotone integer mapping (init must be -inf)
__device__ __forceinline__ void atomicMaxF(float* addr, float v) {
  if (v >= 0.f) atomicMax((int*)addr, __float_as_int(v));
  else          atomicMin((unsigned int*)addr, __float_as_uint(v));
}

__global__ void k_fill(float* __restrict__ p, float v, int count) {
  int i = blockIdx.x * blockDim.x + threadIdx.x;
  if (i < count) p[i] = v;
}

// h1 = x @ W1 ; alpha_src/dst dot products per head. x:[N,2], W1:[2,64]
__global__ void k_l1_transform(const float* __restrict__ x, const float* __restrict__ W1,
                               const float* __restrict__ as_w, const float* __restrict__ ad_w,
                               float* __restrict__ h1, float* __restrict__ as1,
                               float* __restrict__ ad1) {
  int n = blockIdx.x * blockDim.x + threadIdx.x;
  if (n >= NN) return;
  float x0 = x[2 * n], x1 = x[2 * n + 1];
  float s[4] = {0.f, 0.f, 0.f, 0.f}, d[4] = {0.f, 0.f, 0.f, 0.f};
#pragma unroll
  for (int c = 0; c < 64; ++c) {
    float h = fmaf(x0, W1[c], x1 * W1[64 + c]);
    h1[(size_t)n * 64 + c] = h;
    s[c >> 4] = fmaf(h, as_w[c], s[c >> 4]);
    d[c >> 4] = fmaf(h, ad_w[c], d[c >> 4]);
  }
#pragma unroll
  for (int hh = 0; hh < 4; ++hh) { as1[n * 4 + hh] = s[hh]; ad1[n * 4 + hh] = d[hh]; }
}

__global__ void k_edge_max_l1(const int* __restrict__ ei, const float* __restrict__ as1,
                              const float* __restrict__ ad1, float* __restrict__ m1) {
  int e = blockIdx.x * blockDim.x + threadIdx.x;
  if (e >= ETOT) return;
  int s, dd;
  if (e < EE) { s = ei[e]; dd = ei[EE + e]; } else { s = e - EE; dd = s; }
#pragma unroll
  for (int h = 0; h < 4; ++h)
    atomicMaxF(&m1[dd * 4 + h], leaky(as1[s * 4 + h] + ad1[dd * 4 + h]));
}

// one thread per (edge, head): numerator & denominator accumulation
__global__ void k_edge_acc_l1(const int* __restrict__ ei, const float* __restrict__ as1,
                              const float* __restrict__ ad1, const float* __restrict__ m1,
                              const float* __restrict__ h1, float* __restrict__ d1,
                              float* __restrict__ agg1) {
  int gid = blockIdx.x * blockDim.x + threadIdx.x;
  if (gid >= ETOT * 4) return;
  int e = gid >> 2, h = gid & 3;
  int s, dd;
  if (e < EE) { s = ei[e]; dd = ei[EE + e]; } else { s = e - EE; dd = s; }
  float v = leaky(as1[s * 4 + h] + ad1[dd * 4 + h]);
  float p = __expf(v - m1[dd * 4 + h]);
  atomicAdd(&d1[dd * 4 + h], p);
  const float4* hv = (const float4*)(h1 + (size_t)s * 64 + h * 16);
  float* ag = agg1 + (size_t)dd * 64 + h * 16;
#pragma unroll
  for (int i = 0; i < 4; ++i) {
    float4 m = hv[i];
    atomicAdd(ag + 4 * i + 0, p * m.x);
    atomicAdd(ag + 4 * i + 1, p * m.y);
    atomicAdd(ag + 4 * i + 2, p * m.z);
    atomicAdd(ag + 4 * i + 3, p * m.w);
  }
}

// x2 = relu(agg1/denom + b1), one thread per (node,channel)
__global__ void k_finalize_l1(const float* __restrict__ agg1, const float* __restrict__ d1,
                              const float* __restrict__ b1, float* __restrict__ x2) {
  int gid = blockIdx.x * blockDim.x + threadIdx.x;
  if (gid >= NN * 64) return;
  int n = gid >> 6, c = gid & 63;
  float v = agg1[gid] / (d1[n * 4 + (c >> 4)] + 1e-16f) + b1[c];
  x2[gid] = v > 0.f ? v : 0.f;
}

// h2 = x2 @ W2 via V_WMMA_F32_16X16X4_F32, one 16x64x16 tile per wave32.
// A 16x4 f32 layout: lanes 0-15 -> K={0,1} in v0/v1, lanes 16-31 -> K={2,3}
// B 4x16 f32 layout: same K split across lane halves, N = lane&15
__global__ void k_l2_wmma(const float* __restrict__ x2, const float* __restrict__ W2,
                          float* __restrict__ h2) {
  int wave = (blockIdx.x * blockDim.x + threadIdx.x) >> 5;
  int lane = threadIdx.x & 31;
  if (wave >= NTILES) return;  // wave-uniform exit, EXEC stays all-ones for active waves
  int row0 = wave * 16;
  int r = lane & 15;
  int koff = (lane >> 4) * 2;
  v8f acc = {0.f, 0.f, 0.f, 0.f, 0.f, 0.f, 0.f, 0.f};
#pragma unroll
  for (int kb = 0; kb < 16; ++kb) {
    int k = kb * 4 + koff;
    v2f a = *(const v2f*)(x2 + (size_t)(row0 + r) * 64 + k);  // 8B aligned (k even)
    v2f b;
    b.x = W2[(k + 0) * 16 + r];
    b.y = W2[(k + 1) * 16 + r];
    acc = __builtin_amdgcn_wmma_f32_16x16x4_f32(false, a, false, b, (short)0, acc,
                                                false, false);
  }
  int mbase = (lane >> 4) * 8;
#pragma unroll
  for (int v = 0; v < 8; ++v)
    h2[(size_t)(row0 + mbase + v) * 16 + r] = acc[v];
}

__global__ void k_l2_alpha(const float* __restrict__ h2, const float* __restrict__ as_w,
                           const float* __restrict__ ad_w, float* __restrict__ as2,
                           float* __restrict__ ad2) {
  int n = blockIdx.x * blockDim.x + threadIdx.x;
  if (n >= NN) return;
  float s = 0.f, d = 0.f;
#pragma unroll
  for (int c = 0; c < 16; ++c) {
    float hv = h2[(size_t)n * 16 + c];
    s = fmaf(hv, as_w[c], s);
    d = fmaf(hv, ad_w[c], d);
  }
  as2[n] = s; ad2[n] = d;
}

__global__ void k_edge_max_l2(const int* __restrict__ ei, const float* __restrict__ as2,
                              const float* __restrict__ ad2, float* __restrict__ m2) {
  int e = blockIdx.x * blockDim.x + threadIdx.x;
  if (e >= ETOT) return;
  int s, dd;
  if (e < EE) { s = ei[e]; dd = ei[EE + e]; } else { s = e - EE; dd = s; }
  atomicMaxF(&m2[dd], leaky(as2[s] + ad2[dd]));
}

__global__ void k_edge_acc_l2(const int* __restrict__ ei, const float* __restrict__ as2,
                              const float* __restrict__ ad2, const float* __restrict__ m2,
                              const float* __restrict__ h2, float* __restrict__ d2,
                              float* __restrict__ agg2) {
  int e = blockIdx.x * blockDim.x + threadIdx.x;
  if (e >= ETOT) return;
  int s, dd;
  if (e < EE) { s = ei[e]; dd = ei[EE + e]; } else { s = e - EE; dd = s; }
  float p = __expf(leaky(as2[s] + ad2[dd]) - m2[dd]);
  atomicAdd(&d2[dd], p);
  const float4* hv = (const float4*)(h2 + (size_t)s * 16);
  float* ag = agg2 + (size_t)dd * 16;
#pragma unroll
  for (int i = 0; i < 4; ++i) {
    float4 m = hv[i];
    atomicAdd(ag + 4 * i + 0, p * m.x);
    atomicAdd(ag + 4 * i + 1, p * m.y);
    atomicAdd(ag + 4 * i + 2, p * m.z);
    atomicAdd(ag + 4 * i + 3, p * m.w);
  }
}

// out = relu(agg2/denom + b2) @ Wo + bo
__global__ void k_finalize_out(const float* __restrict__ agg2, const float* __restrict__ d2,
                               const float* __restrict__ b2, const float* __restrict__ Wo,
                               const float* __restrict__ bo, float* __restrict__ out) {
  int n = blockIdx.x * blockDim.x + threadIdx.x;
  if (n >= NN) return;
  float den = d2[n] + 1e-16f;
  float acc = 0.f;
#pragma unroll
  for (int c = 0; c < 16; ++c) {
    float v = agg2[(size_t)n * 16 + c] / den + b2[c];
    v = v > 0.f ? v : 0.f;
    acc = fmaf(v, Wo[c], acc);
  }
  out[n] = acc + bo[0];
}

extern "C" void kernel_launch(void* const* d_in, const int* in_sizes, int n_in,
                              void* d_out, int out_size, void* d_ws, size_t ws_size,
                              hipStream_t stream) {
  (void)in_sizes; (void)n_in; (void)out_size; (void)ws_size;
  const float* x      = (const float*)d_in[0];
  const int*   ei     = (const int*)d_in[1];
  const float* W1     = (const float*)d_in[2];
  const float* a_src1 = (const float*)d_in[3];
  const float* a_dst1 = (const float*)d_in[4];
  const float* b1     = (const float*)d_in[5];
  const float* W2     = (const float*)d_in[6];
  const float* a_src2 = (const float*)d_in[7];
  const float* a_dst2 = (const float*)d_in[8];
  const float* b2     = (const float*)d_in[9];
  const float* Wo     = (const float*)d_in[10];
  const float* bo     = (const float*)d_in[11];
  float* out = (float*)d_out;

  float* ws = (float*)d_ws;
  const size_t NS = (size_t)NN;
  float* h1   = ws;               // 64N  (reused as x2 after finalize_l1)
  float* agg1 = ws + 64 * NS;     // 64N
  float* as1  = ws + 128 * NS;    // 4N
  float* ad1  = ws + 132 * NS;    // 4N
  float* m1   = ws + 136 * NS;    // 4N
  float* d1   = ws + 140 * NS;    // 4N
  float* h2   = ws + 144 * NS;    // 16N
  float* agg2 = ws + 160 * NS;    // 16N
  float* as2  = ws + 176 * NS;    // N
  float* ad2  = ws + 177 * NS;    // N
  float* m2   = ws + 178 * NS;    // N
  float* d2   = ws + 179 * NS;    // N  -> total 180N floats = 72 MB

  const int B = 256;
  const float NINF = -__builtin_huge_valf();
  k_fill<<<(4 * NN + B - 1) / B, B, 0, stream>>>(m1, NINF, 4 * NN);
  k_fill<<<(4 * NN + B - 1) / B, B, 0, stream>>>(d1, 0.f, 4 * NN);
  k_fill<<<(64 * NN + B - 1) / B, B, 0, stream>>>(agg1, 0.f, 64 * NN);
  k_fill<<<(NN + B - 1) / B, B, 0, stream>>>(m2, NINF, NN);
  k_fill<<<(NN + B - 1) / B, B, 0, stream>>>(d2, 0.f, NN);
  k_fill<<<(16 * NN + B - 1) / B, B, 0, stream>>>(agg2, 0.f, 16 * NN);

  k_l1_transform<<<(NN + B - 1) / B, B, 0, stream>>>(x, W1, a_src1, a_dst1, h1, as1, ad1);
  k_edge_max_l1<<<(ETOT + B - 1) / B, B, 0, stream>>>(ei, as1, ad1, m1);
  k_edge_acc_l1<<<(ETOT * 4 + B - 1) / B, B, 0, stream>>>(ei, as1, ad1, m1, h1, d1, agg1);
  k_finalize_l1<<<(64 * NN + B - 1) / B, B, 0, stream>>>(agg1, d1, b1, /*x2=*/h1);

  // 4 waves per block, one 16-row tile per wave
  k_l2_wmma<<<(NTILES + 3) / 4, 128, 0, stream>>>(/*x2=*/h1, W2, h2);

  k_l2_alpha<<<(NN + B - 1) / B, B, 0, stream>>>(h2, a_src2, a_dst2, as2, ad2);
  k_edge_max_l2<<<(ETOT + B - 1) / B, B, 0, stream>>>(ei, as2, ad2, m2);
  k_edge_acc_l2<<<(ETOT + B - 1) / B, B, 0, stream>>>(ei, as2, ad2, m2, h2, d2, agg2);
  k_finalize_out<<<(NN + B - 1) / B, B, 0, stream>>>(agg2, d2, b2, Wo, bo, out);
}